// LMOptimizer_2705829396750
// MI455X (gfx1250) — compile-verified
//
#include <hip/hip_runtime.h>
#include <float.h>

typedef __attribute__((ext_vector_type(2))) float v2f;
typedef __attribute__((ext_vector_type(4))) float v4f;
typedef __attribute__((ext_vector_type(8))) float v8f;

#define B_      32
#define M_      307200
#define PB      75          // blocks per batch element
#define ITERS   16          // rows per lane: 75 * 256 * 16 == 307200 exactly

// d_ws float-offsets
#define WS_THETA 0          // committed theta   [32][4]
#define WS_THEV  128        // candidate theta   [32][4] (what lm_pass evaluates)
#define WS_LAMB  256        // lambda            [32]
#define WS_COST  288        // committed cost    [32]
#define WS_STOP  320        // stop flag (0/1)   [32]
#define WS_PART  352        // block partials    [32][75][16]

__device__ __forceinline__ float wred(float v) {
    v += __shfl_xor(v, 1);
    v += __shfl_xor(v, 2);
    v += __shfl_xor(v, 4);
    v += __shfl_xor(v, 8);
    v += __shfl_xor(v, 16);
    return v;
}

// One fused sweep at theta_eval: accumulates per-block
//   q 0..9  : H uniques (00,01,02,03,11,12,13,22,23,33)  with w = huber_d1*conf
//   q 10..13: G = sum w*r*J
//   q 14    : cost = sum huber_loss*a2*conf
__global__ __launch_bounds__(256) void lm_pass(const float* __restrict__ J,
                                               const float* __restrict__ bvec,
                                               const float* __restrict__ conf,
                                               float* __restrict__ ws) {
    __shared__ float plds[8][16];
    const int batch = blockIdx.x / PB;
    const int blk   = blockIdx.x % PB;
    const int tid   = threadIdx.x;
    const int lane  = tid & 31;
    const int wv    = tid >> 5;

    const float* th = ws + WS_THEV + batch * 4;
    const float t0 = th[0], t1 = th[1], t2 = th[2], t3 = th[3];
    const float stopped = ws[WS_STOP + batch];

    const size_t row0 = (size_t)batch * M_ + (size_t)blk * (256 * ITERS);
    const v4f*   __restrict__ jp = (const v4f*)(J + row0 * 4);
    const float* __restrict__ bp = bvec + row0;
    const float* __restrict__ cp = conf + row0;

    float h00=0.f,h01=0.f,h02=0.f,h03=0.f,h11=0.f,h12=0.f,h13=0.f,h22=0.f,h23=0.f,h33=0.f;
    float g0=0.f,g1=0.f,g2=0.f,g3=0.f,cacc=0.f;

    if (stopped == 0.0f) {   // frozen batch entries never affect outputs; skip the sweep
        #pragma unroll 4
        for (int i = 0; i < ITERS; ++i) {
            const int m = i * 256 + tid;
            v4f  jv = __builtin_nontemporal_load(jp + m);   // J: streamed once, keep out of L2
            float bv = bp[m];                               // b/conf: L2-resident across passes
            float cv = cp[m];
            float pred = jv.x * t0;
            pred = fmaf(jv.y, t1, pred);
            pred = fmaf(jv.z, t2, pred);
            pred = fmaf(jv.w, t3, pred);
            float r  = bv - pred;
            float x  = (r * r) * 1.0e4f;                    // r^2 / a^2, a=0.01
            float sx = sqrtf(x + 1e-8f);
            float isx = fmaxf(FLT_EPSILON, 1.0f / sx);
            bool  qd  = (x <= 1.0f);
            float loss = qd ? x : fmaf(2.0f, sx, -1.0f);
            float d1   = qd ? 1.0f : isx;
            cacc = fmaf(loss * 1.0e-4f, cv, cacc);          // loss * a2 * conf
            float w  = d1 * cv;
            float wj0 = w * jv.x, wj1 = w * jv.y, wj2 = w * jv.z, wj3 = w * jv.w;
            float wr  = w * r;
            h00 = fmaf(wj0, jv.x, h00); h01 = fmaf(wj0, jv.y, h01);
            h02 = fmaf(wj0, jv.z, h02); h03 = fmaf(wj0, jv.w, h03);
            h11 = fmaf(wj1, jv.y, h11); h12 = fmaf(wj1, jv.z, h12);
            h13 = fmaf(wj1, jv.w, h13);
            h22 = fmaf(wj2, jv.z, h22); h23 = fmaf(wj2, jv.w, h23);
            h33 = fmaf(wj3, jv.w, h33);
            g0 = fmaf(wr, jv.x, g0); g1 = fmaf(wr, jv.y, g1);
            g2 = fmaf(wr, jv.z, g2); g3 = fmaf(wr, jv.w, g3);
        }
    }

    // intra-wave butterfly reductions (wave32)
    h00 = wred(h00); h01 = wred(h01); h02 = wred(h02); h03 = wred(h03);
    h11 = wred(h11); h12 = wred(h12); h13 = wred(h13);
    h22 = wred(h22); h23 = wred(h23); h33 = wred(h33);
    g0  = wred(g0);  g1  = wred(g1);  g2  = wred(g2);  g3  = wred(g3);
    cacc = wred(cacc);

    if (lane == 0) {
        float* row = plds[wv];
        row[0]=h00; row[1]=h01; row[2]=h02;  row[3]=h03;  row[4]=h11;
        row[5]=h12; row[6]=h13; row[7]=h22;  row[8]=h23;  row[9]=h33;
        row[10]=g0; row[11]=g1; row[12]=g2;  row[13]=g3;  row[14]=cacc;
        row[15]=0.f;
    }
    __syncthreads();

    // Cross-wave reduction via V_WMMA_F32_16X16X4_F32 on wave 0 (full EXEC):
    //   D = ones(16x4) * B(4x16), B[k][q] = wave-k partial q  ->  D[*][q] = sum_k partial_q
    // A is all-ones so the result is invariant to the exact K<->lane mapping of B.
    if (tid < 32) {
        const int col = tid & 15;
        const int k0  = (tid < 16) ? 0 : 2;
        v2f a;  a.x = 1.0f;            a.y = 1.0f;
        v2f b0; b0.x = plds[k0][col];     b0.y = plds[k0 + 1][col];   // waves 0..3
        v2f b1; b1.x = plds[4 + k0][col]; b1.y = plds[5 + k0][col];   // waves 4..7
        v8f c = {};
        c = __builtin_amdgcn_wmma_f32_16x16x4_f32(false, a, false, b0, (short)0, c, false, false);
        c = __builtin_amdgcn_wmma_f32_16x16x4_f32(false, a, false, b1, (short)0, c, false, false);
        if (tid < 16) {
            float* out = ws + WS_PART + ((size_t)batch * PB + blk) * 16;
            out[tid] = c[0];          // D[0][col] = block total for quantity `col`
        }
    }
}

// One lane per batch element: combine 75 partials, LM bookkeeping, damped 4x4 Cholesky.
__global__ void lm_update(float* __restrict__ ws, int first) {
    const int b = threadIdx.x;
    if (b >= B_) return;
    float s[15];
    #pragma unroll
    for (int q = 0; q < 15; ++q) s[q] = 0.f;
    const float* part = ws + WS_PART + (size_t)b * PB * 16;
    for (int blk = 0; blk < PB; ++blk) {
        #pragma unroll
        for (int q = 0; q < 15; ++q) s[q] += part[blk * 16 + q];
    }
    float h00=s[0],h01=s[1],h02=s[2],h03=s[3],h11=s[4],h12=s[5],h13=s[6],h22=s[7],h23=s[8],h33=s[9];
    float g0=s[10],g1=s[11],g2=s[12],g3=s[13];
    float cost_now = s[14];

    float lamb = ws[WS_LAMB + b];
    float stop = ws[WS_STOP + b];
    if (first) {
        ws[WS_COST + b] = cost_now;                       // cost0 at theta0
    } else {
        float prev = ws[WS_COST + b];
        float nl = lamb * ((cost_now > prev) ? 10.f : 0.1f);
        nl = fminf(fmaxf(nl, 1e-6f), 100.f);
        bool conv = fabsf(cost_now - prev) <= (1e-8f + 1e-8f * fabsf(prev));
        if (stop == 0.f) {
            #pragma unroll
            for (int n = 0; n < 4; ++n)
                ws[WS_THETA + b*4 + n] = ws[WS_THEV + b*4 + n];
            ws[WS_LAMB + b] = nl;
            ws[WS_COST + b] = cost_now;
            lamb = nl;
        }
        if (conv) ws[WS_STOP + b] = 1.f;
    }

    // LM damping: H[p][p] += clamp(lambda * H[p][p], EPS_DAMP, inf)
    h00 += fmaxf(lamb * h00, 1e-6f);
    h11 += fmaxf(lamb * h11, 1e-6f);
    h22 += fmaxf(lamb * h22, 1e-6f);
    h33 += fmaxf(lamb * h33, 1e-6f);

    // Cholesky H = L L^T (H is SPD: J^T W J + positive damping)
    float l00 = sqrtf(h00);                  float i00 = 1.f / l00;
    float l10 = h01 * i00, l20 = h02 * i00, l30 = h03 * i00;
    float l11 = sqrtf(h11 - l10 * l10);      float i11 = 1.f / l11;
    float l21 = (h12 - l20 * l10) * i11;
    float l31 = (h13 - l30 * l10) * i11;
    float l22 = sqrtf(h22 - l20 * l20 - l21 * l21);  float i22 = 1.f / l22;
    float l32 = (h23 - l30 * l20 - l31 * l21) * i22;
    float l33 = sqrtf(h33 - l30 * l30 - l31 * l31 - l32 * l32);  float i33 = 1.f / l33;
    // forward: L y = G
    float y0 = g0 * i00;
    float y1 = (g1 - l10 * y0) * i11;
    float y2 = (g2 - l20 * y0 - l21 * y1) * i22;
    float y3 = (g3 - l30 * y0 - l31 * y1 - l32 * y2) * i33;
    // backward: L^T d = y
    float d3 = y3 * i33;
    float d2 = (y2 - l32 * d3) * i22;
    float d1 = (y1 - l21 * d2 - l31 * d3) * i11;
    float d0 = (y0 - l10 * d1 - l20 * d2 - l30 * d3) * i00;

    ws[WS_THEV + b*4 + 0] = ws[WS_THETA + b*4 + 0] + d0;
    ws[WS_THEV + b*4 + 1] = ws[WS_THETA + b*4 + 1] + d1;
    ws[WS_THEV + b*4 + 2] = ws[WS_THETA + b*4 + 2] + d2;
    ws[WS_THEV + b*4 + 3] = ws[WS_THETA + b*4 + 3] + d3;
}

__global__ void lm_init(const float* __restrict__ theta0, float* __restrict__ ws) {
    const int b = threadIdx.x;
    if (b < B_) {
        #pragma unroll
        for (int n = 0; n < 4; ++n) {
            ws[WS_THETA + b*4 + n] = theta0[b*4 + n];
            ws[WS_THEV  + b*4 + n] = theta0[b*4 + n];
        }
        ws[WS_LAMB + b] = 0.1f;
        ws[WS_COST + b] = 0.f;
        ws[WS_STOP + b] = 0.f;
    }
}

__global__ void lm_final(const float* __restrict__ ws, float* __restrict__ out) {
    const int i = threadIdx.x;
    if (i < 128)       out[i] = ws[WS_THETA + i];        // theta (32,4)
    else if (i < 160)  out[i] = ws[WS_COST + (i - 128)]; // cost  (32,)
}

extern "C" void kernel_launch(void* const* d_in, const int* in_sizes, int n_in,
                              void* d_out, int out_size, void* d_ws, size_t ws_size,
                              hipStream_t stream) {
    (void)in_sizes; (void)n_in; (void)out_size; (void)ws_size;
    const float* J      = (const float*)d_in[0];
    const float* bvec   = (const float*)d_in[1];
    const float* conf   = (const float*)d_in[2];
    const float* theta0 = (const float*)d_in[3];
    float* ws  = (float*)d_ws;
    float* out = (float*)d_out;

    lm_init<<<1, 32, 0, stream>>>(theta0, ws);
    // 31 fused sweeps: pass(theta0) produces cost0/H0/G0; then 30 candidate evaluations.
    for (int k = 0; k <= 30; ++k) {
        lm_pass<<<B_ * PB, 256, 0, stream>>>(J, bvec, conf, ws);
        lm_update<<<1, 32, 0, stream>>>(ws, (k == 0) ? 1 : 0);
    }
    lm_final<<<1, 192, 0, stream>>>(ws, out);
}